// GATNet_48790828483155
// MI455X (gfx1250) — compile-verified
//
#include <hip/hip_runtime.h>
#include <hip/hip_bf16.h>

typedef __attribute__((ext_vector_type(2))) float v2f;
typedef __attribute__((ext_vector_type(8))) float v8f;

#define N_NODES 40000
#define N_EDGES 640000
#define N_ETOT  (N_EDGES + N_NODES)
#define N_GRAPH 64
#define DIM0    128
#define HC      256   /* heads * channels for hidden layers */
#define CH      64
#define NHEAD   4
#define SLOPE   0.2f

__device__ __forceinline__ float lrelu(float v) { return v > 0.f ? v : SLOPE * v; }

// Monotone mapping float -> uint so unsigned atomicMax == float max (handles negatives).
__device__ __forceinline__ unsigned fkey(float f) {
    unsigned u = __float_as_uint(f);
    return (u & 0x80000000u) ? ~u : (u | 0x80000000u);
}
__device__ __forceinline__ float unfkey(unsigned u) {
    unsigned b = (u & 0x80000000u) ? (u ^ 0x80000000u) : ~u;
    return __uint_as_float(b);
}

// ---------------------------------------------------------------------------
// Y[n,M] = A[n,K] @ W[K,M] in fp32 via V_WMMA_F32_16X16X4_F32.
// One wave per 32x64 output strip (2 row tiles x 4 col tiles, 8 accumulators):
// per K-step the 4 B fragments (8 scalar loads) feed 8 WMMAs, and the 2 A
// fragments (2 x b64) are each reused across 4 column tiles.
// A frag (16x4 f32, 2 VGPR): lane<16 -> row=lane, K={k,k+1}; lane>=16 -> K={k+2,k+3}.
// B frag (4x16 f32, 2 VGPR): VGPR0 = rows {k | k+2}, VGPR1 = rows {k+1 | k+3}.
// D (16x16 f32, 8 VGPR): VGPR j: lanes0-15 -> row j, lanes16-31 -> row j+8, col=lane&15.
// ---------------------------------------------------------------------------
__global__ void gemm_f32_wmma(const float* __restrict__ A, const float* __restrict__ W,
                              float* __restrict__ Y, int n, int K, int M) {
    int gw = (blockIdx.x * blockDim.x + threadIdx.x) >> 5;
    int stripsN = M >> 6;                       // 64-wide output strips
    int strips  = (n >> 5) * stripsN;           // 32-row strips
    if (gw >= strips) return;
    int rowBase = (gw / stripsN) << 5;
    int colBase = (gw % stripsN) << 6;
    int lane = threadIdx.x & 31;
    int half = lane >> 4;     // 0: K pair {0,1}, 1: K pair {2,3}
    int l15  = lane & 15;

    const float* arow0 = A + (size_t)(rowBase + l15) * K + half * 2;
    const float* arow1 = arow0 + (size_t)16 * K;
    const float* wcol  = W + (size_t)(half * 2) * M + colBase + l15;

    v8f z = {0.f,0.f,0.f,0.f,0.f,0.f,0.f,0.f};
    v8f acc00 = z, acc01 = z, acc02 = z, acc03 = z;   // rows [rowBase, rowBase+16)
    v8f acc10 = z, acc11 = z, acc12 = z, acc13 = z;   // rows [rowBase+16, rowBase+32)
    for (int k = 0; k < K; k += 4) {
        float2 a2u = *reinterpret_cast<const float2*>(arow0 + k);
        float2 a2l = *reinterpret_cast<const float2*>(arow1 + k);
        v2f au; au.x = a2u.x; au.y = a2u.y;
        v2f al; al.x = a2l.x; al.y = a2l.y;
        const float* wk = wcol + (size_t)k * M;
        v2f b0; b0.x = wk[0];   b0.y = wk[M];          // col tile 0
        v2f b1; b1.x = wk[16];  b1.y = wk[M + 16];     // col tile 1
        v2f b2; b2.x = wk[32];  b2.y = wk[M + 32];     // col tile 2
        v2f b3; b3.x = wk[48];  b3.y = wk[M + 48];     // col tile 3
        acc00 = __builtin_amdgcn_wmma_f32_16x16x4_f32(false, au, false, b0, (short)0, acc00, false, false);
        acc01 = __builtin_amdgcn_wmma_f32_16x16x4_f32(false, au, false, b1, (short)0, acc01, false, false);
        acc02 = __builtin_amdgcn_wmma_f32_16x16x4_f32(false, au, false, b2, (short)0, acc02, false, false);
        acc03 = __builtin_amdgcn_wmma_f32_16x16x4_f32(false, au, false, b3, (short)0, acc03, false, false);
        acc10 = __builtin_amdgcn_wmma_f32_16x16x4_f32(false, al, false, b0, (short)0, acc10, false, false);
        acc11 = __builtin_amdgcn_wmma_f32_16x16x4_f32(false, al, false, b1, (short)0, acc11, false, false);
        acc12 = __builtin_amdgcn_wmma_f32_16x16x4_f32(false, al, false, b2, (short)0, acc12, false, false);
        acc13 = __builtin_amdgcn_wmma_f32_16x16x4_f32(false, al, false, b3, (short)0, acc13, false, false);
    }
    float* y0 = Y + (size_t)(rowBase + half * 8) * M + colBase + l15;
    float* y1 = y0 + (size_t)16 * M;
#pragma unroll
    for (int j = 0; j < 8; ++j) {
        y0[(size_t)j * M +  0] = acc00[j];
        y0[(size_t)j * M + 16] = acc01[j];
        y0[(size_t)j * M + 32] = acc02[j];
        y0[(size_t)j * M + 48] = acc03[j];
        y1[(size_t)j * M +  0] = acc10[j];
        y1[(size_t)j * M + 16] = acc11[j];
        y1[(size_t)j * M + 32] = acc12[j];
        y1[(size_t)j * M + 48] = acc13[j];
    }
}

// ---------------------------------------------------------------------------
// Pass B: per-edge GATv2 score + segment max. One wave per edge.
// Lane handles 8 contiguous channels (head = lane>>3); reduce within 8-lane group.
// ---------------------------------------------------------------------------
__global__ void edge_score_max(const float* __restrict__ xl, const float* __restrict__ xr,
                               const int* __restrict__ ei, const float* __restrict__ att,
                               float* __restrict__ score, unsigned* __restrict__ smax) {
    int e = (blockIdx.x * blockDim.x + threadIdx.x) >> 5;
    if (e >= N_ETOT) return;
    int lane = threadIdx.x & 31;
    int src, dst;
    if (e < N_EDGES) { src = ei[e]; dst = ei[N_EDGES + e]; }
    else             { src = dst = e - N_EDGES; }

    const float4* pl = reinterpret_cast<const float4*>(xl + (size_t)src * HC + lane * 8);
    const float4* pr = reinterpret_cast<const float4*>(xr + (size_t)dst * HC + lane * 8);
    const float4* pa = reinterpret_cast<const float4*>(att + lane * 8);

    float s = 0.f;
#pragma unroll
    for (int t = 0; t < 2; ++t) {
        float4 l = pl[t], r = pr[t], a = pa[t];
        s += lrelu(l.x + r.x) * a.x;
        s += lrelu(l.y + r.y) * a.y;
        s += lrelu(l.z + r.z) * a.z;
        s += lrelu(l.w + r.w) * a.w;
    }
    s += __shfl_xor(s, 1, 32);
    s += __shfl_xor(s, 2, 32);
    s += __shfl_xor(s, 4, 32);
    if ((lane & 7) == 0) {
        int h = lane >> 3;
        score[(size_t)e * NHEAD + h] = s;
        atomicMax(&smax[(size_t)dst * NHEAD + h], fkey(s));
    }
}

// Pass C: escore = exp(score - smax[dst]); denom[dst] += escore. Thread per (edge,head).
__global__ void edge_expsum(const int* __restrict__ ei, const unsigned* __restrict__ smax,
                            float* __restrict__ score, float* __restrict__ denom) {
    int idx = blockIdx.x * blockDim.x + threadIdx.x;
    if (idx >= N_ETOT * NHEAD) return;
    int e = idx >> 2, h = idx & 3;
    int dst = (e < N_EDGES) ? ei[N_EDGES + e] : (e - N_EDGES);
    float m  = unfkey(smax[(size_t)dst * NHEAD + h]);
    float es = __expf(score[idx] - m);
    score[idx] = es;
    atomicAdd(&denom[(size_t)dst * NHEAD + h], es);
}

// Pass D: agg[dst] += alpha * xl[src]. One wave per edge, 8 channels/lane.
__global__ void edge_aggregate(const float* __restrict__ xl, const int* __restrict__ ei,
                               const float* __restrict__ escore, const float* __restrict__ denom,
                               float* __restrict__ agg) {
    int e = (blockIdx.x * blockDim.x + threadIdx.x) >> 5;
    if (e >= N_ETOT) return;
    int lane = threadIdx.x & 31;
    int src, dst;
    if (e < N_EDGES) { src = ei[e]; dst = ei[N_EDGES + e]; }
    else             { src = dst = e - N_EDGES; }
    int h = lane >> 3;
    float alpha = escore[(size_t)e * NHEAD + h] /
                  (denom[(size_t)dst * NHEAD + h] + 1e-16f);
    const float4* pl = reinterpret_cast<const float4*>(xl + (size_t)src * HC + lane * 8);
    float* pd = agg + (size_t)dst * HC + lane * 8;
    float4 v0 = pl[0], v1 = pl[1];
    atomicAdd(pd + 0, alpha * v0.x);
    atomicAdd(pd + 1, alpha * v0.y);
    atomicAdd(pd + 2, alpha * v0.z);
    atomicAdd(pd + 3, alpha * v0.w);
    atomicAdd(pd + 4, alpha * v1.x);
    atomicAdd(pd + 5, alpha * v1.y);
    atomicAdd(pd + 6, alpha * v1.z);
    atomicAdd(pd + 7, alpha * v1.w);
}

// Layers 1-2: h = leaky_relu(agg + bias), concat layout (in place).
__global__ void finalize_cat(float* __restrict__ agg, const float* __restrict__ bias) {
    int idx = blockIdx.x * blockDim.x + threadIdx.x;
    if (idx >= N_NODES * HC) return;
    agg[idx] = lrelu(agg[idx] + bias[idx & (HC - 1)]);
}

// Layer 3: mean over heads + bias (no activation).
__global__ void finalize_mean(const float* __restrict__ agg, const float* __restrict__ bias,
                              float* __restrict__ h3) {
    int idx = blockIdx.x * blockDim.x + threadIdx.x;
    if (idx >= N_NODES * CH) return;
    int i = idx >> 6, c = idx & 63;
    const float* p = agg + (size_t)i * HC;
    h3[idx] = 0.25f * (p[c] + p[64 + c] + p[128 + c] + p[192 + c]) + bias[c];
}

// global_max_pool via ordered-key atomicMax.
__global__ void pool_max(const float* __restrict__ h3, const int* __restrict__ batch,
                         unsigned* __restrict__ gmax) {
    int idx = blockIdx.x * blockDim.x + threadIdx.x;
    if (idx >= N_NODES * CH) return;
    int i = idx >> 6, c = idx & 63;
    atomicMax(&gmax[(size_t)batch[i] * CH + c], fkey(h3[idx]));
}

// Decode pooled keys -> g (d_out[128..4223]); logits = g @ Wc + bc (d_out[0..127]).
__global__ void head_out(const unsigned* __restrict__ gmax, const float* __restrict__ Wc,
                         const float* __restrict__ bc, float* __restrict__ out) {
    int idx = blockIdx.x * blockDim.x + threadIdx.x;
    if (idx < N_GRAPH * CH) out[N_GRAPH * 2 + idx] = unfkey(gmax[idx]);
    if (idx < N_GRAPH * 2) {
        int b = idx >> 1, o = idx & 1;
        float s = bc[o];
        for (int c = 0; c < CH; ++c) s += unfkey(gmax[b * CH + c]) * Wc[c * 2 + o];
        out[idx] = s;
    }
}

extern "C" void kernel_launch(void* const* d_in, const int* in_sizes, int n_in,
                              void* d_out, int out_size, void* d_ws, size_t ws_size,
                              hipStream_t stream) {
    (void)in_sizes; (void)n_in; (void)out_size; (void)ws_size;
    const float* x    = (const float*)d_in[0];
    const int*   ei   = (const int*)d_in[1];
    const int*   bat  = (const int*)d_in[2];
    const float* Wl1  = (const float*)d_in[3];
    const float* Wr1  = (const float*)d_in[4];
    const float* att1 = (const float*)d_in[5];
    const float* b1   = (const float*)d_in[6];
    const float* Wl2  = (const float*)d_in[7];
    const float* Wr2  = (const float*)d_in[8];
    const float* att2 = (const float*)d_in[9];
    const float* b2   = (const float*)d_in[10];
    const float* Wl3  = (const float*)d_in[11];
    const float* Wr3  = (const float*)d_in[12];
    const float* att3 = (const float*)d_in[13];
    const float* b3   = (const float*)d_in[14];
    const float* Wc   = (const float*)d_in[15];
    const float* bc   = (const float*)d_in[16];

    char* ws = (char*)d_ws;
    const size_t SZH = (size_t)N_NODES * HC * sizeof(float);      // 40.96 MB
    float*    xl    = (float*)(ws);
    float*    xr    = (float*)(ws + SZH);
    float*    hbuf  = (float*)(ws + 2 * SZH);                     // layer in/out + agg
    float*    h3    = (float*)(ws + 3 * SZH);
    char*     p     = ws + 3 * SZH + (size_t)N_NODES * CH * sizeof(float);
    float*    score = (float*)p;             p += (size_t)N_ETOT * NHEAD * sizeof(float);
    unsigned* smax  = (unsigned*)p;          p += (size_t)N_NODES * NHEAD * sizeof(unsigned);
    float*    denom = (float*)p;             p += (size_t)N_NODES * NHEAD * sizeof(float);
    unsigned* gmax  = (unsigned*)p;

    const int gemmStrips = (N_NODES / 32) * (HC / 64);             // 32x64 per wave
    const int gemmBlocks = (gemmStrips * 32 + 255) / 256;
    const int edgeBlocks = (N_ETOT + 7) / 8;                       // 1 wave per edge
    const int ehBlocks   = (N_ETOT * NHEAD + 255) / 256;
    const int ncBlocks   = (N_NODES * HC + 255) / 256;
    const int n64Blocks  = (N_NODES * CH + 255) / 256;

    auto run_layer = [&](const float* inp, int K, const float* Wl, const float* Wr,
                         const float* att, const float* bias, bool mean_out) {
        gemm_f32_wmma<<<gemmBlocks, 256, 0, stream>>>(inp, Wl, xl, N_NODES, K, HC);
        gemm_f32_wmma<<<gemmBlocks, 256, 0, stream>>>(inp, Wr, xr, N_NODES, K, HC);
        hipMemsetAsync(smax,  0, (size_t)N_NODES * NHEAD * sizeof(unsigned), stream);
        hipMemsetAsync(denom, 0, (size_t)N_NODES * NHEAD * sizeof(float), stream);
        hipMemsetAsync(hbuf,  0, SZH, stream);  // safe: GEMMs already consumed inp(==hbuf)
        edge_score_max<<<edgeBlocks, 256, 0, stream>>>(xl, xr, ei, att, score, smax);
        edge_expsum<<<ehBlocks, 256, 0, stream>>>(ei, smax, score, denom);
        edge_aggregate<<<edgeBlocks, 256, 0, stream>>>(xl, ei, score, denom, hbuf);
        if (!mean_out)
            finalize_cat<<<ncBlocks, 256, 0, stream>>>(hbuf, bias);
        else
            finalize_mean<<<n64Blocks, 256, 0, stream>>>(hbuf, bias, h3);
    };

    run_layer(x,    DIM0, Wl1, Wr1, att1, b1, false);
    run_layer(hbuf, HC,   Wl2, Wr2, att2, b2, false);
    run_layer(hbuf, HC,   Wl3, Wr3, att3, b3, true);

    hipMemsetAsync(gmax, 0, (size_t)N_GRAPH * CH * sizeof(unsigned), stream);
    pool_max<<<n64Blocks, 256, 0, stream>>>(h3, bat, gmax);
    head_out<<<(N_GRAPH * CH + 255) / 256, 256, 0, stream>>>(gmax, Wc, bc, (float*)d_out);
}